// HypConv3_53987738911750
// MI455X (gfx1250) — compile-verified
//
#include <hip/hip_runtime.h>
#include <math.h>

typedef __attribute__((ext_vector_type(2))) float v2f;
typedef __attribute__((ext_vector_type(8))) float v8f;

#define BS    32
#define CIN   64
#define COUT  128
#define HP    66
#define WP    66
#define NSP   4096          // Ho*Wo = 64*64
#define UPITCH 4356         // 66*66

#define C_CURV  0.05f
#define SQC     0.22360679774997896f     // sqrt(0.05)
#define MIN_NORM 1e-15f
#define MAXNORM  (0.996f / SQC)          // (1 - 4e-3)/sqrt(c)

// ---- LDS layout (floats) for the scan kernel ----
#define OFF_ACC 0                         // 16 rows * 4097 (pad row -> distinct banks)
#define ACC_PITCH 4097
#define OFF_U   (16 * ACC_PITCH)          // 4356 (16B aligned: 262208)
#define OFF_W   (OFF_U + UPITCH)          // 16 co * 12 taps (taps>=9 zeroed)
#define OFF_SV2 (OFF_W + 192)             // 16: per-co sum v^2 (reused as sum(acc))
#define OFF_SAV (OFF_SV2 + 16)            // 16: per-co sum acc*v
#define OFF_X2  (OFF_SAV + 16)            // 16: carried ||acc||^2
#define OFF_AS  (OFF_X2 + 16)             // 16: scalar A
#define OFF_BS  (OFF_AS + 16)             // 16: scalar B
#define OFF_BN  (OFF_BS + 16)             // 1 : ||bias||^2
#define SMEM_FLOATS (OFF_BN + 1)
#define SMEM_BYTES  (SMEM_FLOATS * 4)

// ============================================================
// Kernel A: u = logmap0(pad(x)) into [b][c][66*66] scratch
// ============================================================
__global__ void logmap_pad_kernel(const float* __restrict__ x,
                                  float* __restrict__ u) {
    const int c = blockIdx.x, b = blockIdx.y;
    const float* xr = x + ((size_t)(b * CIN + c)) * NSP;
    float* ur = u + ((size_t)(b * CIN + c)) * UPITCH;

    __shared__ float s_n2;
    if (threadIdx.x == 0) s_n2 = 0.f;
    __syncthreads();

    float p = 0.f;
    for (int i = threadIdx.x; i < NSP; i += 256) {
        float v = xr[i];
        p = fmaf(v, v, p);
    }
    #pragma unroll
    for (int o = 16; o > 0; o >>= 1) p += __shfl_down(p, o, 32);
    if ((threadIdx.x & 31) == 0) atomicAdd(&s_n2, p);
    __syncthreads();

    const float n = fmaxf(sqrtf(s_n2), MIN_NORM);
    const float z = fminf(SQC * n, 1.f - 1e-7f);
    const float art = 0.5f * (log1pf(z) - log1pf(-z));
    const float s = art / (SQC * n);

    for (int i = threadIdx.x; i < UPITCH; i += 256) {
        const int hp = i / WP, wp = i - hp * WP;
        float val = 0.f;
        if (hp >= 1 && hp <= 64 && wp >= 1 && wp <= 64)
            val = xr[(hp - 1) * 64 + (wp - 1)] * s;
        ur[i] = val;
    }
}

// ============================================================
// Kernel B: sequential Mobius scan over cin, acc LDS-resident.
// One workgroup = (batch b, 16 output channels). 512 thr = 16 waves.
// Conv tiles via V_WMMA_F32_16X16X4_F32 (K=9 padded to 12; B rows
// for taps 9..11 are staged as zeros, so A gathers are unconditional).
// u tile staged via GLOBAL_LOAD_ASYNC_TO_LDS_B128 (ASYNCcnt).
// ============================================================
__global__ void __launch_bounds__(512, 1)
hyp_scan_kernel(const float* __restrict__ u,       // [BS][CIN][UPITCH]
                const float* __restrict__ weight,  // [COUT][CIN][3][3]
                const float* __restrict__ bias,    // [COUT]
                float* __restrict__ out) {         // [BS][COUT][NSP]
    extern __shared__ float sm[];
    float* accS = sm + OFF_ACC;
    float* uS   = sm + OFF_U;
    float* wS   = sm + OFF_W;

    const int tid  = threadIdx.x;
    const int lane = tid & 31;
    const int wv   = tid >> 5;          // 0..15
    const int half = lane >> 4;         // 0/1 : selects K pair for A/B operands
    const int co   = lane & 15;         // N index (output channel within group)
    const int b = blockIdx.y, coBase = blockIdx.x * 16;

    // ---- init: acc = 0, ||acc||^2 = 0, bias-norm slot = 0 ----
    for (int i = tid; i < 16 * ACC_PITCH; i += 512) accS[i] = 0.f;
    if (tid < 16) sm[OFF_X2 + tid] = 0.f;
    if (tid == 0) sm[OFF_BN] = 0.f;
    __syncthreads();
    if (tid < COUT) { float bv = bias[tid]; atomicAdd(&sm[OFF_BN], bv * bv); }

    const float* ubase = u + (size_t)b * CIN * UPITCH;
    const unsigned int uLds = (unsigned int)(uintptr_t)(void*)uS;

    for (int j = 0; j < CIN; ++j) {
        // ---- async-DMA stage u_j (1089 x b128), stage weights, zero slots ----
        const float* uj = ubase + (size_t)j * UPITCH;
        for (int i = tid; i < UPITCH / 4; i += 512) {
            const unsigned long long ga =
                (unsigned long long)(uintptr_t)(uj + 4 * i);
            const unsigned int la = uLds + 16u * (unsigned int)i;
            asm volatile("global_load_async_to_lds_b128 %0, %1, off"
                         :: "v"(la), "v"(ga) : "memory");
            if (j + 1 < CIN) __builtin_prefetch(uj + UPITCH + 4 * i, 0, 0);
        }
        if (tid < 192) {
            const int cl = tid / 12, tp = tid - cl * 12;
            wS[tid] = (tp < 9)
                ? weight[((size_t)(coBase + cl) * CIN + j) * 9 + tp] : 0.f;
        }
        if (tid < 16) { sm[OFF_SV2 + tid] = 0.f; sm[OFF_SAV + tid] = 0.f; }
        asm volatile("s_wait_asynccnt 0x0" ::: "memory");
        __syncthreads();

        // ---- per-lane B operands (constant across tiles; taps>=9 are 0) ----
        v2f bm[3];
        #pragma unroll
        for (int c = 0; c < 3; ++c) {
            const int kk = 4 * c + 2 * half;
            bm[c].x = wS[co * 12 + kk];
            bm[c].y = wS[co * 12 + kk + 1];
        }

        const int m0   = lane & 15;          // A-matrix M = spatial within tile
        const int arow = co * ACC_PITCH;

        // ================= PASS 1: v via WMMA; row reductions =================
        float pv2 = 0.f, pav = 0.f;
        for (int tile = wv; tile < 256; tile += 16) {
            v8f cm = {};
            const int s0 = tile * 16 + m0;
            const int base = (s0 >> 6) * WP + (s0 & 63);
            #pragma unroll
            for (int c = 0; c < 3; ++c) {
                const int kk = 4 * c + 2 * half;
                const int q0 = (kk * 11) >> 5, r0 = kk - 3 * q0;
                const int k1 = kk + 1;
                const int q1 = (k1 * 11) >> 5, r1 = k1 - 3 * q1;
                v2f am;                       // taps>=9: B row is 0 -> value dont-care
                am.x = uS[base + q0 * WP + r0];
                am.y = uS[base + q1 * WP + r1];
                cm = __builtin_amdgcn_wmma_f32_16x16x4_f32(
                         false, am, false, bm[c], (short)0, cm, false, false);
            }
            const int sbase = tile * 16 + 8 * half;
            #pragma unroll
            for (int r = 0; r < 8; ++r) {
                const float vv = cm[r];
                const float av = accS[arow + sbase + r];
                pv2 = fmaf(vv, vv, pv2);
                pav = fmaf(av, vv, pav);
            }
        }
        atomicAdd(&sm[OFF_SV2 + co], pv2);
        atomicAdd(&sm[OFF_SAV + co], pav);
        __syncthreads();

        // ---- per-co scalars: expmap0 -> project -> mobius_add -> project ----
        if (tid < 16) {
            const float v2 = sm[OFF_SV2 + tid];
            const float av = sm[OFF_SAV + tid];
            const float x2 = sm[OFF_X2 + tid];
            const float vn = fmaxf(sqrtf(v2), MIN_NORM);
            float alpha = tanhf(SQC * vn) / (SQC * vn);     // t = alpha*v
            const float tn = alpha * vn;
            if (tn > MAXNORM) alpha *= MAXNORM / tn;        // project(t)
            const float y2 = alpha * alpha * v2;
            const float xy = alpha * av;
            const float den = fmaxf(
                1.f + 2.f * C_CURV * xy + C_CURV * C_CURV * x2 * y2, MIN_NORM);
            float A = (1.f + 2.f * C_CURV * xy + C_CURV * y2) / den;
            float B = alpha * (1.f - C_CURV * x2) / den;    // acc' = A*acc + B*v
            float n2 = A * A * x2 + 2.f * A * B * av + B * B * v2;
            const float n = fmaxf(sqrtf(n2), MIN_NORM);
            if (n > MAXNORM) {                               // project(acc')
                const float pj = MAXNORM / n;
                A *= pj; B *= pj; n2 = MAXNORM * MAXNORM;
            }
            sm[OFF_AS + tid] = A;
            sm[OFF_BS + tid] = B;
            sm[OFF_X2 + tid] = n2;
        }
        __syncthreads();

        const float As = sm[OFF_AS + co];
        const float Bs = sm[OFF_BS + co];

        // ================= PASS 2: recompute v; acc = A*acc + B*v =============
        for (int tile = wv; tile < 256; tile += 16) {
            v8f cm = {};
            const int s0 = tile * 16 + m0;
            const int base = (s0 >> 6) * WP + (s0 & 63);
            #pragma unroll
            for (int c = 0; c < 3; ++c) {
                const int kk = 4 * c + 2 * half;
                const int q0 = (kk * 11) >> 5, r0 = kk - 3 * q0;
                const int k1 = kk + 1;
                const int q1 = (k1 * 11) >> 5, r1 = k1 - 3 * q1;
                v2f am;
                am.x = uS[base + q0 * WP + r0];
                am.y = uS[base + q1 * WP + r1];
                cm = __builtin_amdgcn_wmma_f32_16x16x4_f32(
                         false, am, false, bm[c], (short)0, cm, false, false);
            }
            const int sbase = tile * 16 + 8 * half;
            #pragma unroll
            for (int r = 0; r < 8; ++r) {
                const int ai = arow + sbase + r;
                accS[ai] = fmaf(As, accS[ai], Bs * cm[r]);
            }
        }
        __syncthreads();
    }

    // ======== EPILOGUE: out = project(mobius_add(acc, expmap0(bias))) ========
    if (tid < 16) sm[OFF_SV2 + tid] = 0.f;   // reuse as per-co sum(acc)
    __syncthreads();
    {
        const int co_t = tid & 15;
        float ps = 0.f;
        for (int s = (tid >> 4); s < NSP; s += 32)
            ps += accS[co_t * ACC_PITCH + s];
        atomicAdd(&sm[OFF_SV2 + co_t], ps);
    }
    __syncthreads();
    if (tid < 16) {
        const float sacc = sm[OFF_SV2 + tid];
        const float x2   = sm[OFF_X2 + tid];
        const float bn   = fmaxf(sqrtf(sm[OFF_BN]), MIN_NORM);
        const float bh   = tanhf(SQC * bn) / (SQC * bn) * bias[coBase + tid];
        const float y2 = (float)NSP * bh * bh;
        const float xy = bh * sacc;
        const float den = fmaxf(
            1.f + 2.f * C_CURV * xy + C_CURV * C_CURV * x2 * y2, MIN_NORM);
        float A  = (1.f + 2.f * C_CURV * xy + C_CURV * y2) / den;
        float Bc = (1.f - C_CURV * x2) * bh / den;      // out = A*acc + Bc
        const float n2 = A * A * x2 + 2.f * A * Bc * sacc + (float)NSP * Bc * Bc;
        const float n  = fmaxf(sqrtf(n2), MIN_NORM);
        if (n > MAXNORM) { const float pj = MAXNORM / n; A *= pj; Bc *= pj; }
        sm[OFF_AS + tid] = A;
        sm[OFF_BS + tid] = Bc;
    }
    __syncthreads();

    float* outb = out + ((size_t)b * COUT + coBase) * NSP;
    for (int i = tid; i < 16 * NSP; i += 512) {
        const int cl = i >> 12, s = i & 4095;
        outb[(size_t)cl * NSP + s] =
            fmaf(sm[OFF_AS + cl], accS[cl * ACC_PITCH + s], sm[OFF_BS + cl]);
    }
}

// ============================================================
extern "C" void kernel_launch(void* const* d_in, const int* in_sizes, int n_in,
                              void* d_out, int out_size, void* d_ws, size_t ws_size,
                              hipStream_t stream) {
    const float* x      = (const float*)d_in[0];  // [32,64,64,64]
    const float* weight = (const float*)d_in[1];  // [128,64,3,3]
    const float* bias   = (const float*)d_in[2];  // [128]
    float* out = (float*)d_out;                   // [32,128,64,64]
    float* u   = (float*)d_ws;                    // [32,64,4356] = 35.7 MB

    (void)in_sizes; (void)n_in; (void)out_size; (void)ws_size;

    // allow ~281 KB dynamic LDS (CDNA5 WGP: 320 KB)
    hipFuncSetAttribute(reinterpret_cast<const void*>(hyp_scan_kernel),
                        hipFuncAttributeMaxDynamicSharedMemorySize, SMEM_BYTES);

    logmap_pad_kernel<<<dim3(CIN, BS), 256, 0, stream>>>(x, u);
    hyp_scan_kernel<<<dim3(COUT / 16, BS), 512, SMEM_BYTES, stream>>>(
        u, weight, bias, out);
}